// ROIPooling2d_47991964566222
// MI455X (gfx1250) — compile-verified
//
#include <hip/hip_runtime.h>
#include <cstdint>

// ROI max pooling (Fast R-CNN / torchvision roi_pool semantics), fp32.
// x: (B=4, C=256, H=50, W=50), rois: (M,5) = [b, x1, y1, x2, y2] in image coords,
// out: (M, C, 7, 7). spatial_scale = 1/16, round-half-even box projection.

#define PH_ 7
#define PW_ 7

constexpr float kScale = 0.0625f;
constexpr int kC = 256;
constexpr int kH = 50;
constexpr int kW = 50;
constexpr int kBlock = 64; // 2 wave32 waves

typedef __attribute__((address_space(1))) int g_int;
typedef __attribute__((address_space(3))) int l_int;

#if __has_builtin(__builtin_amdgcn_global_load_async_to_lds_b32)
#define HAVE_ASYNC_LDS 1
#else
#define HAVE_ASYNC_LDS 0
#endif

__global__ __launch_bounds__(kBlock) void ROIPooling2d_roi_max_kernel(
    const float* __restrict__ x, const float* __restrict__ rois,
    float* __restrict__ out) {
  // Max staged window: 50x50 floats = 10 KB LDS.
  __shared__ float tile[kH * kW];

  const int c = blockIdx.x;   // channel (adjacent blocks touch contiguous planes)
  const int m = blockIdx.y;   // roi
  const int tid = threadIdx.x;

  // --- ROI projection (same float ops as reference) ---
  const float r0 = rois[m * 5 + 0];
  const int b  = (int)r0;
  const int x1 = (int)rintf(rois[m * 5 + 1] * kScale); // rint = round-half-even (jnp.round)
  const int y1 = (int)rintf(rois[m * 5 + 2] * kScale);
  const int x2 = (int)rintf(rois[m * 5 + 3] * kScale);
  const int y2 = (int)rintf(rois[m * 5 + 4] * kScale);
  const int roi_w = max(x2 - x1 + 1, 1);
  const int roi_h = max(y2 - y1 + 1, 1);
  const float bin_w = (float)roi_w / (float)PW_; // true IEEE f32 divide
  const float bin_h = (float)roi_h / (float)PH_;

  // Union of all bin ranges, computed with the SAME float expressions the bins
  // use (ceil(7*bin) may exceed roi_dim by 1 ulp-rounding; must not truncate).
  const int hs0 = min(max(y1, 0), kH);
  const int ws0 = min(max(x1, 0), kW);
  const int he0 = min(max((int)ceilf((float)PH_ * bin_h) + y1, 0), kH);
  const int we0 = min(max((int)ceilf((float)PW_ * bin_w) + x1, 0), kW);
  const int win_h = he0 - hs0;
  const int win_w = we0 - ws0;
  const int total = win_h * win_w;

  const float* __restrict__ plane =
      x + ((size_t)b * kC + (size_t)c) * (size_t)(kH * kW);

  // --- Stage window into LDS (async global->LDS on CDNA5) ---
  for (int i = tid; i < total; i += kBlock) {
    const int rr = i / win_w;
    const int cc = i - rr * win_w;
    const float* gp = plane + (hs0 + rr) * kW + (ws0 + cc);
#if HAVE_ASYNC_LDS
    __builtin_amdgcn_global_load_async_to_lds_b32(
        (g_int*)(uintptr_t)gp, (l_int*)&tile[i], 0, 0);
#else
    tile[i] = __builtin_nontemporal_load(gp);
#endif
  }

#if HAVE_ASYNC_LDS
#if __has_builtin(__builtin_amdgcn_s_wait_asynccnt)
  __builtin_amdgcn_s_wait_asynccnt(0);
#else
  asm volatile("s_wait_asynccnt 0" ::: "memory");
#endif
#endif
  __syncthreads();

  // --- 49 bins, one per thread; reads from LDS; coalesced 49-float store ---
  if (tid < PH_ * PW_) {
    const int ph = tid / PW_;
    const int pw = tid - ph * PW_;
    const int hs = min(max((int)floorf((float)ph * bin_h) + y1, 0), kH);
    const int he = min(max((int)ceilf((float)(ph + 1) * bin_h) + y1, 0), kH);
    const int wcs = min(max((int)floorf((float)pw * bin_w) + x1, 0), kW);
    const int wce = min(max((int)ceilf((float)(pw + 1) * bin_w) + x1, 0), kW);

    float v = -__builtin_inff();
    for (int h = hs; h < he; ++h) {
      const float* row = &tile[(h - hs0) * win_w];
      for (int w = wcs; w < wce; ++w) {
        v = fmaxf(v, row[w - ws0]);
      }
    }
    if (hs >= he || wcs >= wce) v = 0.0f; // empty bin -> 0 (matches reference)

    out[((size_t)m * kC + (size_t)c) * (PH_ * PW_) + (size_t)tid] = v;
  }
}

extern "C" void kernel_launch(void* const* d_in, const int* in_sizes, int n_in,
                              void* d_out, int out_size, void* d_ws, size_t ws_size,
                              hipStream_t stream) {
  (void)n_in; (void)out_size; (void)d_ws; (void)ws_size;
  const float* x    = (const float*)d_in[0];
  const float* rois = (const float*)d_in[1];
  float* out        = (float*)d_out;

  const int M = in_sizes[1] / 5; // (M,5) rois

  dim3 grid(kC, M);
  ROIPooling2d_roi_max_kernel<<<grid, kBlock, 0, stream>>>(x, rois, out);
}